// NCJT_RxUE_22600117911708
// MI455X (gfx1250) — compile-verified
//
#include <hip/hip_runtime.h>

// Problem constants (match reference module constants)
#define B_    64
#define FFT_  512
#define S_    14
#define R_    4
#define T_    4
#define E_    484   // len(arange(14,499)) - 1 (DC removed)
#define D_    12    // 14 symbols - 2 pilots
#define P_    6     // Alamouti pairs
#define M_    16    // 16-QAM points
#define MB_   4     // bits per symbol

// Flat float offsets into d_out (outputs concatenated in return order):
// yd (B,E,48) f32 | gains (B,E,6) f32 | nvar (1) | y (B,E,12) c64 | h_hat (B,E,4,2) c64
#define YD_OFF   0ull
#define G_OFF    1486848ull              // 64*484*48
#define NV_OFF   1672704ull              // + 64*484*6
#define Y_OFF    1672705ull              // + 1
#define HH_OFF   2416129ull              // + 64*484*12*2

typedef __attribute__((ext_vector_type(2))) float v2f;
typedef __attribute__((ext_vector_type(8))) float v8f;

#if defined(__has_builtin)
#if __has_builtin(__builtin_amdgcn_wmma_f32_16x16x4_f32)
#define USE_WMMA 1
#endif
#endif
#ifndef USE_WMMA
#define USE_WMMA 0
#endif

// One thread per (b, e, pair). 64*484*6 = 185856 = 726 * 256 exactly -> EXEC
// is always all-ones (required for WMMA).
__global__ __launch_bounds__(256) void ncjt_combine_demap(
    const float* __restrict__ ry,     // (B,FFT,S,R,1) complex64 as float pairs
    const float* __restrict__ hf,     // (B,E,S,R,T)  complex64
    const float* __restrict__ cpts,   // (16,) complex64
    const float* __restrict__ cbits,  // (16,4) f32
    const int*   __restrict__ dsyms,  // (12,)
    const int*   __restrict__ effsc,  // (E,)
    float* __restrict__ out)
{
    __shared__ float2 ybuf[8][64];    // per-wave y values (re,im)
    __shared__ int    ibuf[8][64];    // per-wave demap indices

    const int tid = blockIdx.x * 256 + threadIdx.x;
    const int p   = tid % P_;
    const int be  = tid / P_;
    const int e   = be % E_;
    const int b   = be / E_;

    const int s0 = dsyms[2 * p];
    const int s1 = dsyms[2 * p + 1];
    const int f  = effsc[e];

    // ---- load ry for both symbols of the pair: 4 complex each (32B, aligned)
    float q0[8], q1[8];
    {
        const size_t rybase = (size_t)(b * FFT_ + f) * S_;
        const float4* a0 = (const float4*)(ry + (rybase + (size_t)s0) * (R_ * 2));
        const float4* a1 = (const float4*)(ry + (rybase + (size_t)s1) * (R_ * 2));
        *(float4*)(q0)     = a0[0];
        *(float4*)(q0 + 4) = a0[1];
        *(float4*)(q1)     = a1[0];
        *(float4*)(q1 + 4) = a1[1];
    }

    // ---- Alamouti combine over R=4 antennas
    float y1r = 0.f, y1i = 0.f, y2r = 0.f, y2i = 0.f, gg = 0.f;
    {
        const size_t hfb = ((size_t)b * E_ + e) * S_;
        const size_t h0b = (hfb + (size_t)s0) * (R_ * T_ * 2);
        const size_t h1b = (hfb + (size_t)s1) * (R_ * T_ * 2);
        #pragma unroll
        for (int r = 0; r < R_; ++r) {
            // t layout per float4: (t.re, t.im, t+1.re, t+1.im)
            float4 a0 = *(const float4*)(hf + h0b + r * 8);      // s0: t0,t1
            float4 a1 = *(const float4*)(hf + h0b + r * 8 + 4);  // s0: t2,t3
            float4 c0 = *(const float4*)(hf + h1b + r * 8);      // s1: t0,t1
            float4 c1 = *(const float4*)(hf + h1b + r * 8 + 4);  // s1: t2,t3
            // h1 = 0.5*(s0.t0+s1.t0) + 0.5*(s0.t2+s1.t2); h2 uses t1,t3
            float h1re = 0.5f * ((a0.x + c0.x) + (a1.x + c1.x));
            float h1im = 0.5f * ((a0.y + c0.y) + (a1.y + c1.y));
            float h2re = 0.5f * ((a0.z + c0.z) + (a1.z + c1.z));
            float h2im = 0.5f * ((a0.w + c0.w) + (a1.w + c1.w));
            float r1re = q0[2 * r], r1im = q0[2 * r + 1];
            float r2re = q1[2 * r], r2im = q1[2 * r + 1];
            // y1 += conj(h1)*r1 + h2*conj(r2)
            y1r += h1re * r1re + h1im * r1im + h2re * r2re + h2im * r2im;
            y1i += h1re * r1im - h1im * r1re + h2im * r2re - h2re * r2im;
            // y2 += conj(h2)*r1 - h1*conj(r2)
            y2r += h2re * r1re + h2im * r1im - h1re * r2re - h1im * r2im;
            y2i += h2re * r1im - h2im * r1re - h1im * r2re + h1re * r2im;
            gg  += h1re * h1re + h1im * h1im + h2re * h2re + h2im * h2im;
        }
    }
    y1r /= gg; y1i /= gg; y2r /= gg; y2i /= gg;

    // ---- continuous outputs
    out[G_OFF + (size_t)be * P_ + p] = gg;
    {
        const size_t yo = Y_OFF + ((size_t)be * D_ + 2 * p) * 2;
        out[yo + 0] = y1r; out[yo + 1] = y1i;
        out[yo + 2] = y2r; out[yo + 3] = y2i;
    }

    // ---- 16-QAM hard demap
    const int lane = threadIdx.x & 31;
    const int wv   = threadIdx.x >> 5;
    ybuf[wv][lane * 2 + 0] = make_float2(y1r, y1i);
    ybuf[wv][lane * 2 + 1] = make_float2(y2r, y2i);
    __syncthreads();

#if USE_WMMA
    // D = A x B with A = constellation (16x4), B = y-values (4x16):
    //   A rows m:   [c_m.re, c_m.im, -0.5|c_m|^2, 0]   (lanes0-15: K0/K1, lanes16-31: K2/K3)
    //   B cols n:   [y_n.re, y_n.im, 1, 0]^T           (v0 = {K0|K2}, v1 = {K1|K3})
    //   D[m,n] = y_n.re*c_m.re + y_n.im*c_m.im - 0.5|c_m|^2  (argmax_m == argmin_m d2)
    // D layout: lane l (<16): dm[j] = D[j, l]; lane l+16: dm[j] = D[j+8, l].
    // => lane l holds scores of points 0-7 for y_l, lane l+16 points 8-15 for same y.
    const int  cl = lane & 15;
    const bool lo = lane < 16;
    const float cre = cpts[cl * 2 + 0];
    const float cim = cpts[cl * 2 + 1];
    v2f am;
    am.x = lo ? cre : (-0.5f * (cre * cre + cim * cim));
    am.y = lo ? cim : 0.0f;
    const int pbase = lo ? 0 : 8;     // point-row offset held by this lane half
    #pragma unroll
    for (int g4 = 0; g4 < 4; ++g4) {
        float2 yv = ybuf[wv][g4 * 16 + cl];
        v2f bm;
        bm.x = lo ? yv.x : 1.0f;
        bm.y = lo ? yv.y : 0.0f;
        v8f acc = {};
        v8f dm = __builtin_amdgcn_wmma_f32_16x16x4_f32(
            false, am, false, bm, (short)0, acc, false, false);
        // per-lane scan over 8 point scores (strict > keeps lowest index on ties)
        float s = dm[0];
        int   n = pbase;
        #pragma unroll
        for (int j = 1; j < 8; ++j) {
            bool t = dm[j] > s;
            s = t ? dm[j] : s;
            n = t ? (pbase + j) : n;
        }
        // merge point-halves 0-7 vs 8-15 for the same y (tie -> lower index)
        float ps = __shfl_xor(s, 16, 32);
        int   pn = __shfl_xor(n, 16, 32);
        bool take = (ps > s) || ((ps == s) && (pn < n));
        n = take ? pn : n;
        if (lo) ibuf[wv][g4 * 16 + cl] = n;
    }
#else
    // scalar fallback: direct d2 scan (first-min semantics)
    {
        float yy[2][2] = {{y1r, y1i}, {y2r, y2i}};
        for (int q = 0; q < 2; ++q) {
            int best = 0;
            float bd = 3.4e38f;
            for (int k = 0; k < M_; ++k) {
                float dr = yy[q][0] - cpts[k * 2 + 0];
                float di = yy[q][1] - cpts[k * 2 + 1];
                float d2 = dr * dr + di * di;
                if (d2 < bd) { bd = d2; best = k; }
            }
            ibuf[wv][lane * 2 + q] = best;
        }
    }
#endif
    __syncthreads();

    #pragma unroll
    for (int q = 0; q < 2; ++q) {
        int idx = ibuf[wv][lane * 2 + q];
        float4 bits = *(const float4*)(cbits + (size_t)idx * MB_);
        *(float4*)(out + YD_OFF + ((size_t)be * D_ + 2 * p + q) * MB_) = bits;
    }
}

// One thread per (b, e, r). 64*484*4 = 123904 = 484 * 256 exactly.
__global__ __launch_bounds__(256) void ncjt_hhat(
    const float* __restrict__ he,      // (B,FFT,R,2) complex64
    const float* __restrict__ ltf,     // (1,E,1,1) f32
    const float* __restrict__ pshift,  // (1,E,1,2) complex64
    const int*   __restrict__ effsc,   // (E,)
    float* __restrict__ out)
{
    const int tid = blockIdx.x * 256 + threadIdx.x;
    const int r = tid & 3;
    const int e = (tid >> 2) % E_;
    const int b = tid / (E_ * R_);
    const int f = effsc[e];
    const float lv = ltf[e];

    // he_ltf[b,f,r,0..1]: 2 complex = 16B aligned
    const float4 hv = *(const float4*)(he + ((size_t)(b * FFT_ + f) * R_ + r) * 4);
    const float h0re = hv.x * lv, h0im = hv.y * lv;
    const float h1re = hv.z * lv, h1im = hv.w * lv;
    // h @ [[1,1],[-1,1]]: col0 = h0 - h1, col1 = h0 + h1
    const float a0re = h0re - h1re, a0im = h0im - h1im;
    const float a1re = h0re + h1re, a1im = h0im + h1im;
    const float4 pv = *(const float4*)(pshift + (size_t)e * 4);
    const size_t o = HH_OFF + (((size_t)b * E_ + e) * R_ + r) * 4;
    out[o + 0] = a0re * pv.x - a0im * pv.y;
    out[o + 1] = a0re * pv.y + a0im * pv.x;
    out[o + 2] = a1re * pv.z - a1im * pv.w;
    out[o + 3] = a1re * pv.w + a1im * pv.z;

    if (tid == 0) out[NV_OFF] = 0.002f;  // nvar
}

extern "C" void kernel_launch(void* const* d_in, const int* in_sizes, int n_in,
                              void* d_out, int out_size, void* d_ws, size_t ws_size,
                              hipStream_t stream) {
    const float* ry  = (const float*)d_in[0];  // ry_noisy  c64
    const float* he  = (const float*)d_in[1];  // he_ltf    c64
    const float* hf  = (const float*)d_in[2];  // h_freq_ns3 c64
    const float* ltf = (const float*)d_in[3];  // ltf_ref   f32
    const float* ps  = (const float*)d_in[4];  // pshift    c64
    const float* cp  = (const float*)d_in[5];  // const_points c64
    const float* cb  = (const float*)d_in[6];  // const_bits f32
    const int*   dsy = (const int*)d_in[7];    // data_syms
    const int*   eff = (const int*)d_in[8];    // eff_sc
    float* out = (float*)d_out;

    const int n1 = B_ * E_ * P_;   // 185856 = 726*256
    ncjt_combine_demap<<<n1 / 256, 256, 0, stream>>>(ry, hf, cp, cb, dsy, eff, out);

    const int n2 = B_ * E_ * R_;   // 123904 = 484*256
    ncjt_hhat<<<n2 / 256, 256, 0, stream>>>(he, ltf, ps, eff, out);
}